// GCN_reg_e_69483980915281
// MI455X (gfx1250) — compile-verified
//
#include <hip/hip_runtime.h>
#include <float.h>

typedef float v2f __attribute__((ext_vector_type(2)));
typedef float v8f __attribute__((ext_vector_type(8)));

#define N_NODES  100000
#define N_EDGES  1600000
#define F_IN     32
#define HID      128
#define N_GRAPHS 2048
#define BOND_DIM 10

// ---------- monotonic float <-> uint transform (branchless segment-max) ----------
// nonneg float: u = bits | 0x80000000 ; negative float: u = ~bits
// => unsigned ordering of u matches float ordering. Single atomicMax(u32), no divergence.
__device__ __forceinline__ unsigned f2ord(float f) {
    unsigned b = __float_as_uint(f);
    unsigned mask = (unsigned)(((int)b) >> 31) | 0x80000000u;
    return b ^ mask;
}
__device__ __forceinline__ float ord2f(unsigned u) {
    unsigned mask = (~(unsigned)(((int)u) >> 31)) | 0x80000000u;
    return __uint_as_float(u ^ mask);
}
#define ORD_NEG_FLT_MAX 0x00800000u   // f2ord(-FLT_MAX)

__global__ void fill_f32(float* __restrict__ p, float v, int n) {
    int i = blockIdx.x * blockDim.x + threadIdx.x;
    if (i < n) p[i] = v;
}
__global__ void fill_u32(unsigned* __restrict__ p, unsigned v, int n) {
    int i = blockIdx.x * blockDim.x + threadIdx.x;
    if (i < n) p[i] = v;
}

// ---------- per-edge scalar weight: w[e] = edge_a[e,:] . bond_W + bond_b ----------
__global__ void edge_weight_k(const float* __restrict__ edge_a,
                              const float* __restrict__ bond_W,
                              const float* __restrict__ bond_b,
                              float* __restrict__ w) {
    int e = blockIdx.x * blockDim.x + threadIdx.x;   // grid sized exactly E
    const float* a = edge_a + e * BOND_DIM;
    float s = bond_b[0];
#pragma unroll
    for (int d = 0; d < BOND_DIM; ++d) s += a[d] * bond_W[d];
    w[e] = s;
}

// ---------- per-graph node counts ----------
__global__ void count_k(const int* __restrict__ batch, float* __restrict__ cnt) {
    int i = blockIdx.x * blockDim.x + threadIdx.x;
    if (i < N_NODES) atomicAdd(&cnt[batch[i]], 1.0f);
}

// ---------- scatter conv1: agg[tgt] += w[e]*x[src], F=32, 8 threads/edge (float4) ----------
__global__ void scatter1_k(const float* __restrict__ x, const float* __restrict__ w,
                           const int* __restrict__ ei, float* __restrict__ agg) {
    int tid = blockIdx.x * blockDim.x + threadIdx.x;  // E*8 threads exactly
    int e  = tid >> 3;
    int fq = (tid & 7) << 2;
    int src = ei[e];
    int tgt = ei[N_EDGES + e];
    float we = w[e];
    const float4 xv = *(const float4*)&x[src * F_IN + fq];
    float* dst = &agg[tgt * F_IN + fq];
    atomicAdd(dst + 0, we * xv.x);
    atomicAdd(dst + 1, we * xv.y);
    atomicAdd(dst + 2, we * xv.z);
    atomicAdd(dst + 3, we * xv.w);
}

// ---------- scatter conv2: agg[tgt] += w[e]*hr[src], F=128, 32 threads/edge ----------
__global__ void scatter2_k(const float* __restrict__ hr, const float* __restrict__ w,
                           const int* __restrict__ ei, float* __restrict__ agg) {
    int tid = blockIdx.x * blockDim.x + threadIdx.x;  // E*32 threads exactly
    int e  = tid >> 5;
    int fq = (tid & 31) << 2;
    int src = ei[e];
    int tgt = ei[N_EDGES + e];
    float we = w[e];
    const float4 hv = *(const float4*)&hr[src * HID + fq];
    float* dst = &agg[tgt * HID + fq];
    atomicAdd(dst + 0, we * hv.x);
    atomicAdd(dst + 1, we * hv.y);
    atomicAdd(dst + 2, we * hv.z);
    atomicAdd(dst + 3, we * hv.w);
}

// ---------- dense1 (WMMA f32): h1 = agg@rel1^T + b + x@root1^T ; fused pool1 + relu store ----------
// block = 256 thr (8 waves). blockIdx.x = node tile (16 nodes). wave = output-feature tile.
__global__ void dense1_k(const float* __restrict__ agg, const float* __restrict__ x,
                         const float* __restrict__ relW, const float* __restrict__ relB,
                         const float* __restrict__ rootW, const int* __restrict__ batch,
                         float* __restrict__ hr,
                         unsigned* __restrict__ pmax, float* __restrict__ psum) {
    __shared__ float As[16 * F_IN];
    __shared__ float Xs[16 * F_IN];
    const int m0 = blockIdx.x * 16;
    for (int i = threadIdx.x; i < 16 * F_IN; i += 256) {
        int r = i >> 5, c = i & 31;
        As[i] = agg[(m0 + r) * F_IN + c];
        Xs[i] = x[(m0 + r) * F_IN + c];
    }
    __syncthreads();

    const int wave = threadIdx.x >> 5;         // N-tile: features wave*16..+15
    const int lane = threadIdx.x & 31;
    const int m  = lane & 15;                  // A row within tile
    const int hi = lane >> 4;                  // K sub-slot select
    const int n  = lane & 15;                  // B/C column

    v8f c = {};
    const float* W1 = relW  + (wave * 16 + n) * F_IN;
    const float* W2 = rootW + (wave * 16 + n) * F_IN;
#pragma unroll
    for (int kk = 0; kk < F_IN; kk += 4) {
        int k0 = kk + hi * 2;
        v2f a = *(const v2f*)&As[m * F_IN + k0];
        v2f b = *(const v2f*)&W1[k0];
        c = __builtin_amdgcn_wmma_f32_16x16x4_f32(false, a, false, b, (short)0, c, false, false);
    }
#pragma unroll
    for (int kk = 0; kk < F_IN; kk += 4) {
        int k0 = kk + hi * 2;
        v2f a = *(const v2f*)&Xs[m * F_IN + k0];
        v2f b = *(const v2f*)&W2[k0];
        c = __builtin_amdgcn_wmma_f32_16x16x4_f32(false, a, false, b, (short)0, c, false, false);
    }

    const int feat = wave * 16 + n;
    const float bias = relB[feat];
#pragma unroll
    for (int r = 0; r < 8; ++r) {
        int node = m0 + r + 8 * hi;            // C/D layout: VGPR r -> M=r (+8 for hi lanes)
        float val = c[r] + bias;               // pre-relu h1 (pooled)
        int g = batch[node];
        atomicMax(&pmax[g * HID + feat], f2ord(val));   // branchless float max
        atomicAdd(&psum[g * HID + feat], val);
        hr[node * HID + feat] = fmaxf(val, 0.0f);       // relu(h1) for conv2
    }
}

// ---------- dense2 (WMMA f32): h2 = agg@rel2^T + b + hr@root2^T ; pool2 only (h2 never stored) ----------
__global__ void dense2_k(const float* __restrict__ agg, const float* __restrict__ hr,
                         const float* __restrict__ relW, const float* __restrict__ relB,
                         const float* __restrict__ rootW, const int* __restrict__ batch,
                         unsigned* __restrict__ pmax, float* __restrict__ psum) {
    __shared__ float As[16 * HID];
    __shared__ float Hs[16 * HID];
    const int m0 = blockIdx.x * 16;
    for (int i = threadIdx.x; i < 16 * HID; i += 256) {
        int r = i >> 7, c = i & 127;
        As[i] = agg[(m0 + r) * HID + c];
        Hs[i] = hr[(m0 + r) * HID + c];
    }
    __syncthreads();

    const int wave = threadIdx.x >> 5;
    const int lane = threadIdx.x & 31;
    const int m  = lane & 15;
    const int hi = lane >> 4;
    const int n  = lane & 15;

    v8f c = {};
    const float* W1 = relW  + (wave * 16 + n) * HID;
    const float* W2 = rootW + (wave * 16 + n) * HID;
#pragma unroll 8
    for (int kk = 0; kk < HID; kk += 4) {
        int k0 = kk + hi * 2;
        v2f a = *(const v2f*)&As[m * HID + k0];
        v2f b = *(const v2f*)&W1[k0];
        c = __builtin_amdgcn_wmma_f32_16x16x4_f32(false, a, false, b, (short)0, c, false, false);
    }
#pragma unroll 8
    for (int kk = 0; kk < HID; kk += 4) {
        int k0 = kk + hi * 2;
        v2f a = *(const v2f*)&Hs[m * HID + k0];
        v2f b = *(const v2f*)&W2[k0];
        c = __builtin_amdgcn_wmma_f32_16x16x4_f32(false, a, false, b, (short)0, c, false, false);
    }

    const int feat = wave * 16 + n;
    const float bias = relB[feat];
#pragma unroll
    for (int r = 0; r < 8; ++r) {
        int node = m0 + r + 8 * hi;
        float val = c[r] + bias;
        int g = batch[node];
        atomicMax(&pmax[g * HID + feat], f2ord(val));
        atomicAdd(&psum[g * HID + feat], val);
    }
}

// ---------- final MLP per graph: z=[mx1+mx2 || (sm1+sm2)/cnt]; relu(z@lin1^T+b1); @lin2^T+b2 ----------
__global__ void final_k(const unsigned* __restrict__ p1max, const float* __restrict__ p1sum,
                        const unsigned* __restrict__ p2max, const float* __restrict__ p2sum,
                        const float* __restrict__ cnt,
                        const float* __restrict__ lin1W, const float* __restrict__ lin1b,
                        const float* __restrict__ lin2W, const float* __restrict__ lin2b,
                        float* __restrict__ out) {
    __shared__ float z[2 * HID];
    __shared__ float red[HID];
    const int g = blockIdx.x;
    const int t = threadIdx.x;                 // 128 threads
    const float cd = fmaxf(cnt[g], 1.0f);
    z[t]       = ord2f(p1max[g * HID + t]) + ord2f(p2max[g * HID + t]);
    z[t + HID] = (p1sum[g * HID + t] + p2sum[g * HID + t]) / cd;
    __syncthreads();
    float s = lin1b[t];
    const float* Wrow = lin1W + t * (2 * HID);
#pragma unroll 8
    for (int j = 0; j < 2 * HID; ++j) s += z[j] * Wrow[j];
    s = fmaxf(s, 0.0f);
    red[t] = s * lin2W[t];
    __syncthreads();
    for (int off = 64; off > 0; off >>= 1) {
        if (t < off) red[t] += red[t + off];
        __syncthreads();
    }
    if (t == 0) out[g] = red[0] + lin2b[0];
}

// ---------- launch ----------
extern "C" void kernel_launch(void* const* d_in, const int* in_sizes, int n_in,
                              void* d_out, int out_size, void* d_ws, size_t ws_size,
                              hipStream_t stream) {
    const float* x       = (const float*)d_in[0];
    const float* edge_a  = (const float*)d_in[1];
    const float* bond_W  = (const float*)d_in[2];
    const float* bond_b  = (const float*)d_in[3];
    const float* rel1_W  = (const float*)d_in[4];
    const float* rel1_b  = (const float*)d_in[5];
    const float* root1_W = (const float*)d_in[6];
    const float* rel2_W  = (const float*)d_in[7];
    const float* rel2_b  = (const float*)d_in[8];
    const float* root2_W = (const float*)d_in[9];
    const float* lin1_W  = (const float*)d_in[10];
    const float* lin1_b  = (const float*)d_in[11];
    const float* lin2_W  = (const float*)d_in[12];
    const float* lin2_b  = (const float*)d_in[13];
    const int*   ei      = (const int*)d_in[14];
    const int*   batch   = (const int*)d_in[15];
    float* out = (float*)d_out;

    // workspace layout (floats / u32, 4B elements)
    float* ws = (float*)d_ws;
    float*    w_    = ws;                                    // E
    float*    agg_  = w_    + N_EDGES;                       // N*128 (reused both convs)
    float*    hr_   = agg_  + (size_t)N_NODES * HID;         // N*128
    unsigned* p1mx_ = (unsigned*)(hr_ + (size_t)N_NODES * HID); // G*128 (ordered-u32)
    float*    p1sm_ = (float*)(p1mx_ + N_GRAPHS * HID);
    unsigned* p2mx_ = (unsigned*)(p1sm_ + N_GRAPHS * HID);
    float*    p2sm_ = (float*)(p2mx_ + N_GRAPHS * HID);
    float*    cnt_  = p2sm_ + N_GRAPHS * HID;                // G

    const int T = 256;
    // init
    fill_f32<<<(N_NODES * F_IN + T - 1) / T, T, 0, stream>>>(agg_, 0.0f, N_NODES * F_IN);
    fill_u32<<<(N_GRAPHS * HID + T - 1) / T, T, 0, stream>>>(p1mx_, ORD_NEG_FLT_MAX, N_GRAPHS * HID);
    fill_f32<<<(N_GRAPHS * HID + T - 1) / T, T, 0, stream>>>(p1sm_, 0.0f, N_GRAPHS * HID);
    fill_u32<<<(N_GRAPHS * HID + T - 1) / T, T, 0, stream>>>(p2mx_, ORD_NEG_FLT_MAX, N_GRAPHS * HID);
    fill_f32<<<(N_GRAPHS * HID + T - 1) / T, T, 0, stream>>>(p2sm_, 0.0f, N_GRAPHS * HID);
    fill_f32<<<(N_GRAPHS + T - 1) / T, T, 0, stream>>>(cnt_, 0.0f, N_GRAPHS);

    edge_weight_k<<<N_EDGES / T, T, 0, stream>>>(edge_a, bond_W, bond_b, w_);
    count_k<<<(N_NODES + T - 1) / T, T, 0, stream>>>(batch, cnt_);

    // conv1
    scatter1_k<<<(N_EDGES * 8) / T, T, 0, stream>>>(x, w_, ei, agg_);
    dense1_k<<<N_NODES / 16, T, 0, stream>>>(agg_, x, rel1_W, rel1_b, root1_W, batch,
                                             hr_, p1mx_, p1sm_);
    // conv2
    fill_f32<<<((size_t)N_NODES * HID + T - 1) / T, T, 0, stream>>>(agg_, 0.0f, N_NODES * HID);
    scatter2_k<<<((size_t)N_EDGES * 32) / T, T, 0, stream>>>(hr_, w_, ei, agg_);
    dense2_k<<<N_NODES / 16, T, 0, stream>>>(agg_, hr_, rel2_W, rel2_b, root2_W, batch,
                                             p2mx_, p2sm_);
    // readout
    final_k<<<N_GRAPHS, HID, 0, stream>>>(p1mx_, p1sm_, p2mx_, p2sm_, cnt_,
                                          lin1_W, lin1_b, lin2_W, lin2_b, out);
}